// HetGeoAware_20667382628456
// MI455X (gfx1250) — compile-verified
//
#include <hip/hip_runtime.h>
#include <hip/hip_bf16.h>

typedef float v2f __attribute__((ext_vector_type(2)));
typedef float v8f __attribute__((ext_vector_type(8)));

#define N_NODES 8192
#define HDIM    128
#define NT      (N_NODES / 16)   // 512 j-tiles
#define NPAD    132              // padded row stride for 128-wide tiles (16B-aligned, bank-spread)
#define MPAD    20               // padded row stride for 16-wide mask tile
#define COS_TH  0.8f

// ---- CDNA5 async global->LDS helpers (ASYNCcnt path) -----------------------
__device__ __forceinline__ unsigned lds_addr_of(const void* p) {
    // addrspacecast generic -> LDS(as3); as3 pointers are 32-bit LDS offsets
    return (unsigned)(unsigned long long)(__attribute__((address_space(3))) const void*)p;
}

__device__ __forceinline__ void async_copy16(const float* g, float* lds) {
    const unsigned           dst = lds_addr_of(lds);
    const unsigned long long src = (unsigned long long)g;
    asm volatile("global_load_async_to_lds_b128 %0, %1, off"
                 :: "v"(dst), "v"(src) : "memory");
}

__device__ __forceinline__ void wait_async0() {
    asm volatile("s_wait_asynccnt 0x0" ::: "memory");
}

// stage one 16x128 f32 tile (row stride NPAD in LDS) with async-LDS DMA;
// 256 threads x 2 chunks of 16B each
__device__ __forceinline__ void async_stage_tile(float* dstLds, const float* srcGlobal, int tid) {
    #pragma unroll
    for (int q = 0; q < 2; ++q) {
        const int idx = tid + q * 256;       // 0..511
        const int r   = idx >> 5;
        const int c4  = (idx & 31) * 4;
        async_copy16(srcGlobal + r * HDIM + c4, dstLds + r * NPAD + c4);
    }
}

// ---------------------------------------------------------------------------
// K1: row-wise L2 normalize. One block (128 threads) per row.
// ---------------------------------------------------------------------------
__global__ __launch_bounds__(128) void k1_normalize(const float* __restrict__ x,
                                                    float* __restrict__ xn,
                                                    float* __restrict__ normv) {
    const int row = blockIdx.x;
    const int t   = threadIdx.x;          // 0..127 == feature index
    const float v = x[row * HDIM + t];
    float s = v * v;
    #pragma unroll
    for (int m = 16; m >= 1; m >>= 1) s += __shfl_xor(s, m, 32);
    __shared__ float red[4];
    const int wid = t >> 5, lid = t & 31;
    if (lid == 0) red[wid] = s;
    __syncthreads();
    const float tot = red[0] + red[1] + red[2] + red[3];
    const float inv = rsqrtf(tot);
    xn[row * HDIM + t] = v * inv;
    if (t == 0) normv[row] = tot * inv;   // = sqrt(tot)
}

// ---------------------------------------------------------------------------
// K2: degree + adjacency bitmask, double-buffered async j-tile staging.
// ---------------------------------------------------------------------------
__global__ __launch_bounds__(256) void k2_degree(const float* __restrict__ xn,
                                                 const float* __restrict__ normv,
                                                 unsigned short* __restrict__ mask,
                                                 float* __restrict__ dinv,
                                                 float* __restrict__ wscale) {
    __shared__ __attribute__((aligned(16))) float sA[128 * NPAD];     // block's 128 i-rows
    __shared__ __attribute__((aligned(16))) float sB[2][16 * NPAD];   // double-buffered j-tile

    const int tid   = threadIdx.x;
    const int wave  = tid >> 5;
    const int lane  = tid & 31;
    const int iBase = blockIdx.x * 128;

    // stage the 128 i-rows of xn into LDS (synchronous, once)
    for (int idx = tid; idx < 128 * 32; idx += 256) {
        const int r  = idx >> 5;
        const int c4 = (idx & 31) * 4;
        *(float4*)(sA + r * NPAD + c4) = *(const float4*)(xn + (iBase + r) * HDIM + c4);
    }

    const int m     = lane & 15;           // WMMA A/B lane row/col
    const int khalf = (lane >> 4) * 2;     // K sub-offset per lane half
    const float* aRow = sA + (wave * 16 + m) * NPAD;

    int cntLo[8], cntHi[8];
    #pragma unroll
    for (int r = 0; r < 8; ++r) { cntLo[r] = 0; cntHi[r] = 0; }

    // prologue: async-fill buffer 0 with j-tile 0
    async_stage_tile(sB[0], xn, tid);

    for (int jt = 0; jt < NT; ++jt) {
        const int cur = jt & 1;
        wait_async0();
        __syncthreads();                      // tile `jt` ready; prev compute done
        if (jt + 1 < NT)                      // hide next tile's DMA behind compute
            async_stage_tile(sB[cur ^ 1], xn + (jt + 1) * 16 * HDIM, tid);

        const float* bRow = sB[cur] + m * NPAD;
        v8f c = {0.f, 0.f, 0.f, 0.f, 0.f, 0.f, 0.f, 0.f};
        #pragma unroll
        for (int k = 0; k < HDIM; k += 4) {
            const v2f a = *(const v2f*)(aRow + k + khalf);
            const v2f b = *(const v2f*)(bRow + k + khalf);
            c = __builtin_amdgcn_wmma_f32_16x16x4_f32(false, a, false, b,
                                                      (short)0, c, false, false);
        }
        #pragma unroll
        for (int r = 0; r < 8; ++r) {
            const unsigned int bal = __builtin_amdgcn_ballot_w32(c[r] > COS_TH);
            cntLo[r] += __popc(bal & 0xffffu);
            cntHi[r] += __popc(bal >> 16);
            if (lane == 0) {
                mask[(iBase + wave * 16 + r)     * NT + jt] = (unsigned short)(bal & 0xffffu);
                mask[(iBase + wave * 16 + r + 8) * NT + jt] = (unsigned short)(bal >> 16);
            }
        }
    }

    if (lane == 0) {
        #pragma unroll
        for (int r = 0; r < 8; ++r) {
            const int rowLo = iBase + wave * 16 + r;
            const int rowHi = rowLo + 8;
            const float dLo = rsqrtf((float)(cntLo[r] > 1 ? cntLo[r] : 1));
            const float dHi = rsqrtf((float)(cntHi[r] > 1 ? cntHi[r] : 1));
            dinv[rowLo] = dLo;                    dinv[rowHi] = dHi;
            wscale[rowLo] = dLo * normv[rowLo];   wscale[rowHi] = dHi * normv[rowHi];
        }
    }
}

// ---------------------------------------------------------------------------
// K4: normalized aggregation h = dinv_i * sum_j A_ij * wscale_j * xn_j,
//     double-buffered async j-tile staging + mask-expansion WMMA GEMM.
// ---------------------------------------------------------------------------
__global__ __launch_bounds__(256) void k4_aggregate(const float* __restrict__ xn,
                                                    const unsigned short* __restrict__ mask,
                                                    const float* __restrict__ dinv,
                                                    const float* __restrict__ wscale,
                                                    float* __restrict__ hbuf) {
    __shared__ __attribute__((aligned(16))) float sB[2][16 * NPAD];   // j-tile of xn (dbl-buf)
    __shared__ __attribute__((aligned(16))) float sM[8 * 16 * MPAD];  // per-wave masked 16x16

    const int tid   = threadIdx.x;
    const int wave  = tid >> 5;
    const int lane  = tid & 31;
    const int iBase = blockIdx.x * 128;
    const int nloc  = lane & 15;
    const int half  = lane >> 4;
    const int khalf = half * 2;
    const int myRow0 = iBase + wave * 16;
    float* myM = sM + wave * 16 * MPAD;

    v8f acc[8];
    #pragma unroll
    for (int nt = 0; nt < 8; ++nt)
        acc[nt] = (v8f){0.f, 0.f, 0.f, 0.f, 0.f, 0.f, 0.f, 0.f};

    async_stage_tile(sB[0], xn, tid);

    for (int jt = 0; jt < NT; ++jt) {
        const int cur = jt & 1;
        wait_async0();
        __syncthreads();
        if (jt + 1 < NT)
            async_stage_tile(sB[cur ^ 1], xn + (jt + 1) * 16 * HDIM, tid);

        // expand mask bits * wscale_j into wave-local 16x16 operand
        const float wsc = wscale[jt * 16 + nloc];
        #pragma unroll
        for (int r = 0; r < 8; ++r) {
            const int row = r + 8 * half;
            const unsigned int mv = mask[(myRow0 + row) * NT + jt];
            myM[row * MPAD + nloc] = ((mv >> nloc) & 1u) ? wsc : 0.0f;
        }
        __syncthreads();   // order sM stores vs. A-operand loads

        // P(16x128) += M(16x16) @ xn_jtile(16x128), K=16 in 4 WMMA steps
        const float* sBc = sB[cur];
        #pragma unroll
        for (int nt = 0; nt < 8; ++nt) {
            #pragma unroll
            for (int kk = 0; kk < 16; kk += 4) {
                const v2f a = *(const v2f*)(myM + nloc * MPAD + kk + khalf);
                v2f b;
                b.x = sBc[(kk + khalf)     * NPAD + nt * 16 + nloc];
                b.y = sBc[(kk + khalf + 1) * NPAD + nt * 16 + nloc];
                acc[nt] = __builtin_amdgcn_wmma_f32_16x16x4_f32(false, a, false, b,
                                                                (short)0, acc[nt], false, false);
            }
        }
    }

    float dR[8];
    #pragma unroll
    for (int r = 0; r < 8; ++r) dR[r] = dinv[myRow0 + r + 8 * half];
    #pragma unroll
    for (int nt = 0; nt < 8; ++nt)
        #pragma unroll
        for (int r = 0; r < 8; ++r)
            hbuf[(myRow0 + r + 8 * half) * HDIM + nt * 16 + nloc] = acc[nt][r] * dR[r];
}

// ---------------------------------------------------------------------------
// K5: out = relu(h @ W + b). Wave handles 16 rows x 128 cols; bias-init C.
// ---------------------------------------------------------------------------
__global__ __launch_bounds__(256) void k5_dense(const float* __restrict__ hbuf,
                                                const float* __restrict__ W,
                                                const float* __restrict__ bias,
                                                float* __restrict__ out) {
    const int tid   = threadIdx.x;
    const int wave  = tid >> 5;
    const int lane  = tid & 31;
    const int iBase = blockIdx.x * 128;
    const int m     = lane & 15;
    const int nloc  = lane & 15;
    const int half  = lane >> 4;
    const int khalf = half * 2;

    v8f acc[8];
    #pragma unroll
    for (int nt = 0; nt < 8; ++nt) {
        const float bv = bias[nt * 16 + nloc];
        #pragma unroll
        for (int r = 0; r < 8; ++r) acc[nt][r] = bv;
    }

    const float* aRow = hbuf + (iBase + wave * 16 + m) * HDIM;
    for (int k = 0; k < HDIM; k += 4) {
        const v2f a = *(const v2f*)(aRow + k + khalf);
        #pragma unroll
        for (int nt = 0; nt < 8; ++nt) {
            v2f b;
            b.x = W[(k + khalf)     * HDIM + nt * 16 + nloc];
            b.y = W[(k + khalf + 1) * HDIM + nt * 16 + nloc];
            acc[nt] = __builtin_amdgcn_wmma_f32_16x16x4_f32(false, a, false, b,
                                                            (short)0, acc[nt], false, false);
        }
    }

    #pragma unroll
    for (int nt = 0; nt < 8; ++nt)
        #pragma unroll
        for (int r = 0; r < 8; ++r) {
            const float v = acc[nt][r];
            out[(iBase + wave * 16 + r + 8 * half) * HDIM + nt * 16 + nloc] = v > 0.f ? v : 0.f;
        }
}

// ---------------------------------------------------------------------------
extern "C" void kernel_launch(void* const* d_in, const int* in_sizes, int n_in,
                              void* d_out, int out_size, void* d_ws, size_t ws_size,
                              hipStream_t stream) {
    (void)in_sizes; (void)n_in; (void)out_size; (void)ws_size;
    const size_t NH = (size_t)N_NODES * HDIM;

    char* ws = (char*)d_ws;
    float*          xn    = (float*)ws;                              // 4 MB
    float*          hbuf  = (float*)(ws + (4u << 20));               // 4 MB
    unsigned short* maskA = (unsigned short*)(ws + (8u << 20));      // 8 MB
    float*          normv = (float*)(ws + (16u << 20));              // 32 KB
    float*          dinv  = normv + N_NODES;                         // 32 KB
    float*          wsc   = dinv + N_NODES;                          // 32 KB

    const float* feats[2] = {(const float*)d_in[0], (const float*)d_in[1]};
    const float* Wmat[2]  = {(const float*)d_in[2], (const float*)d_in[4]};
    const float* bvec[2]  = {(const float*)d_in[3], (const float*)d_in[5]};
    float* out = (float*)d_out;

    for (int g = 0; g < 2; ++g) {
        k1_normalize<<<N_NODES, 128, 0, stream>>>(feats[g], xn, normv);
        k2_degree  <<<N_NODES / 128, 256, 0, stream>>>(xn, normv, maskA, dinv, wsc);
        k4_aggregate<<<N_NODES / 128, 256, 0, stream>>>(xn, maskA, dinv, wsc, hbuf);
        k5_dense   <<<N_NODES / 128, 256, 0, stream>>>(hbuf, Wmat[g], bvec[g], out + (size_t)g * NH);
    }
}